// Hard_mining_JointsMSELoss_11312943858183
// MI455X (gfx1250) — compile-verified
//
#include <hip/hip_runtime.h>

// Hard-mining JointsMSELoss for B=64, J=17, H=W=128 (f32 in, scalar f32 out).
// Memory-bound single-pass streaming reduction; cross-lane sums done with
// v_wmma_f32_16x16x4_f32 (B = ones), max with wave32 shuffle tree.

#define JOINTS 17
#define BATCH  64
#define HW     16384            // 128*128
#define NBLK   (BATCH * JOINTS) // 1088 contiguous chunks of HW floats

typedef __attribute__((ext_vector_type(2))) float v2f;
typedef __attribute__((ext_vector_type(8))) float v8f;

// Reduce 64 per-lane partials (2 per lane, wave32) to a full-wave sum.
// A[16x4] = the 64 partials (lane L<16: A[L][0..1]=a0,a1; lane L>=16: A[L-16][2..3]).
// B[4x16] = ones  ->  D[r][n] = rowsum(r).  d[0..7] per lane covers rows
// {0..7} (lanes 0-15) or {8..15} (lanes 16-31); one xor-16 shuffle finishes it.
__device__ __forceinline__ float wave_sum_wmma(float a0, float a1) {
    v2f A; A[0] = a0; A[1] = a1;
    v2f Bv; Bv[0] = 1.0f; Bv[1] = 1.0f;
    v8f C = {};
    v8f D = __builtin_amdgcn_wmma_f32_16x16x4_f32(
        /*neg_a=*/false, A, /*neg_b=*/false, Bv,
        /*c_mod=*/(short)0, C, /*reuse_a=*/false, /*reuse_b=*/false);
    float s = ((D[0] + D[1]) + (D[2] + D[3])) + ((D[4] + D[5]) + (D[6] + D[7]));
    s += __shfl_xor(s, 16, 32);
    return s;
}

__device__ __forceinline__ float wave_max(float m) {
#pragma unroll
    for (int off = 16; off >= 1; off >>= 1)
        m = fmaxf(m, __shfl_xor(m, off, 32));
    return m;
}

// One block per (b,j) chunk of 16384 contiguous floats.  blockIdx.x == b*J + j.
__global__ void __launch_bounds__(256)
partial_kernel(const float* __restrict__ pred, const float* __restrict__ gt,
               float* __restrict__ p_se, float* __restrict__ p_cnt,
               float* __restrict__ p_mx) {
    const int blk = blockIdx.x;
    const int tid = threadIdx.x;

    const float4* o4 = (const float4*)pred + (size_t)blk * (HW / 4);
    const float4* t4 = (const float4*)gt   + (size_t)blk * (HW / 4);

    const float NEG_INF = -__builtin_inff();
    float se0 = 0.f, se1 = 0.f;   // squared error over positives (2 accs -> WMMA A)
    float cn0 = 0.f, cn1 = 0.f;   // positive counts as f32 (exact, < 2^24)
    float mx  = NEG_INF;          // max pred where gt == 0

#pragma unroll 4
    for (int i = tid; i < HW / 4; i += 256) {
        float4 p = o4[i];
        float4 g = t4[i];

        float dx = p.x - g.x, dy = p.y - g.y, dz = p.z - g.z, dw = p.w - g.w;
        bool px = g.x > 0.f, py = g.y > 0.f, pz = g.z > 0.f, pw = g.w > 0.f;

        se0 += (px ? dx * dx : 0.f) + (py ? dy * dy : 0.f);
        se1 += (pz ? dz * dz : 0.f) + (pw ? dw * dw : 0.f);
        cn0 += (px ? 1.f : 0.f) + (py ? 1.f : 0.f);
        cn1 += (pz ? 1.f : 0.f) + (pw ? 1.f : 0.f);

        mx = fmaxf(mx, g.x == 0.f ? p.x : NEG_INF);
        mx = fmaxf(mx, g.y == 0.f ? p.y : NEG_INF);
        mx = fmaxf(mx, g.z == 0.f ? p.z : NEG_INF);
        mx = fmaxf(mx, g.w == 0.f ? p.w : NEG_INF);
    }

    // Wave-level reduction (uniform control flow here: EXEC is all-1s for WMMA).
    float wse = wave_sum_wmma(se0, se1);
    float wcn = wave_sum_wmma(cn0, cn1);
    float wmx = wave_max(mx);

    __shared__ float s_se[8], s_cn[8], s_mx[8];
    const int wave = tid >> 5, lane = tid & 31;
    if (lane == 0) { s_se[wave] = wse; s_cn[wave] = wcn; s_mx[wave] = wmx; }
    __syncthreads();

    if (tid == 0) {
        float a = 0.f, c = 0.f, m = NEG_INF;
#pragma unroll
        for (int w = 0; w < 8; ++w) {
            a += s_se[w]; c += s_cn[w]; m = fmaxf(m, s_mx[w]);
        }
        const int j = blk % JOINTS, b = blk / JOINTS;
        const int slot = j * BATCH + b;   // group partials by joint
        p_se[slot] = a; p_cnt[slot] = c; p_mx[slot] = m;
    }
}

// One block, 17 waves: wave j reduces its 64 per-(b,j) partials and computes
// loss_j = sum_se/cnt + max_neg^2; thread 0 averages the 17 joint losses.
__global__ void __launch_bounds__(JOINTS * 32)
finalize_kernel(const float* __restrict__ p_se, const float* __restrict__ p_cnt,
                const float* __restrict__ p_mx, float* __restrict__ out) {
    const int tid  = threadIdx.x;
    const int j    = tid >> 5;
    const int lane = tid & 31;
    const int base = j * BATCH;

    float se = p_se[base + lane]  + p_se[base + 32 + lane];
    float cn = p_cnt[base + lane] + p_cnt[base + 32 + lane];
    float mx = fmaxf(p_mx[base + lane], p_mx[base + 32 + lane]);

    se = wave_sum_wmma(se, 0.f);
    cn = wave_sum_wmma(cn, 0.f);
    mx = wave_max(mx);

    __shared__ float s_loss[JOINTS];
    if (lane == 0) {
        float pos_loss = (cn > 0.f) ? (se / cn) : 0.f;
        s_loss[j] = pos_loss + mx * mx;   // neg_gt is exactly 0
    }
    __syncthreads();

    if (tid == 0) {
        float t = 0.f;
#pragma unroll
        for (int jj = 0; jj < JOINTS; ++jj) t += s_loss[jj];
        out[0] = t / (float)JOINTS;
    }
}

extern "C" void kernel_launch(void* const* d_in, const int* in_sizes, int n_in,
                              void* d_out, int out_size, void* d_ws, size_t ws_size,
                              hipStream_t stream) {
    const float* pred = (const float*)d_in[0];   // "output" [B,J,H,W] f32
    const float* gt   = (const float*)d_in[1];   // "target" [B,J,H,W] f32

    float* ws    = (float*)d_ws;                 // 3 * 1088 floats = 13 KB
    float* p_se  = ws;
    float* p_cnt = ws + NBLK;
    float* p_mx  = ws + 2 * NBLK;

    partial_kernel<<<NBLK, 256, 0, stream>>>(pred, gt, p_se, p_cnt, p_mx);
    finalize_kernel<<<1, JOINTS * 32, 0, stream>>>(p_se, p_cnt, p_mx, (float*)d_out);
}